// QuantizedCNN_62981400429093
// MI455X (gfx1250) — compile-verified
//
#include <hip/hip_runtime.h>
#include <hip/hip_bf16.h>

typedef __attribute__((ext_vector_type(2))) float v2f;
typedef __attribute__((ext_vector_type(8))) float v8f;

// Q8.8 fake-quant: round-to-nearest-even (matches jnp.round), clamp, rescale.
__device__ __forceinline__ float fpq(float v) {
    float q = rintf(v * 256.0f);
    q = fminf(fmaxf(q, -32768.0f), 32767.0f);
    return q * 0.00390625f;  // /256
}

__device__ __forceinline__ v8f wmma4(v2f a, v2f b, v8f c) {
    // V_WMMA_F32_16X16X4_F32: full-fp32 matrix op, 8-arg form
    return __builtin_amdgcn_wmma_f32_16x16x4_f32(
        false, a, false, b, (short)0, c, false, false);
}

// ---------------------------------------------------------------------------
// Kernel 1: conv1 (1->16ch, 3x3, pad 1) + fpq + relu + maxpool2 + fpq, fused.
// One thread per pooled output element [4096,16,14,14].
// ---------------------------------------------------------------------------
__global__ __launch_bounds__(256) void conv1_pool_kernel(
    const float* __restrict__ x, const float* __restrict__ w,
    const float* __restrict__ bias, float* __restrict__ h1) {
    int idx = blockIdx.x * blockDim.x + threadIdx.x;
    const int total = 4096 * 16 * 14 * 14;
    if (idx >= total) return;
    int px = idx % 14;
    int py = (idx / 14) % 14;
    int c  = (idx / 196) % 16;
    int b  = idx / (196 * 16);

    float wk[9];
#pragma unroll
    for (int i = 0; i < 9; ++i) wk[i] = w[c * 9 + i];
    float bc = bias[c];
    const float* xb = x + b * 784;

    float best = -3.4e38f;
#pragma unroll
    for (int dy = 0; dy < 2; ++dy) {
#pragma unroll
        for (int dx = 0; dx < 2; ++dx) {
            int oy = 2 * py + dy, ox = 2 * px + dx;
            float acc = bc;
#pragma unroll
            for (int ky = 0; ky < 3; ++ky) {
                int iy = oy + ky - 1;
#pragma unroll
                for (int kx = 0; kx < 3; ++kx) {
                    int ix = ox + kx - 1;
                    float xv = (iy >= 0 && iy < 28 && ix >= 0 && ix < 28)
                                   ? xb[iy * 28 + ix] : 0.0f;
                    acc = fmaf(xv, wk[ky * 3 + kx], acc);
                }
            }
            best = fmaxf(best, fmaxf(fpq(acc), 0.0f));
        }
    }
    h1[idx] = fpq(best);  // qmaxpool2's fpq (identity on quantized data)
}

// ---------------------------------------------------------------------------
// Kernel 2: conv2 as implicit GEMM with WMMA f32.
// M = 4096*196 = 802816 rows (b,oy,ox), K = 144 (c,ky,kx), N = 32 channels.
// Block = 128 thr (4 waves): stages A[64x144] (im2col) + B[144x32] in LDS,
// each wave computes a 16x16 tile for both N-halves. Fused bias+fpq+relu.
// ---------------------------------------------------------------------------
__global__ __launch_bounds__(128) void conv2_wmma_kernel(
    const float* __restrict__ h1, const float* __restrict__ w2,
    const float* __restrict__ b2, float* __restrict__ h2) {
    __shared__ float As[64 * 148];  // stride 148 avoids bank conflicts
    __shared__ float Bs[144 * 33];
    int tid = threadIdx.x;
    int m_base = blockIdx.x * 64;

    // Stage im2col A tile
    for (int i = tid; i < 64 * 144; i += 128) {
        int r = i / 144, k = i - (i / 144) * 144;
        int g = m_base + r;
        int b = g / 196;
        int p = g - b * 196;
        int oy = p / 14;
        int ox = p - oy * 14;
        int c = k / 9;
        int t = k - c * 9;
        int ky = t / 3;
        int kx = t - ky * 3;
        int iy = oy + ky - 1, ix = ox + kx - 1;
        float v = 0.0f;
        if (iy >= 0 && iy < 14 && ix >= 0 && ix < 14)
            v = h1[((b * 16 + c) * 14 + iy) * 14 + ix];
        As[r * 148 + k] = v;
    }
    // Stage weights: B[k][n] = w2[n][k]  (OIHW flat: n*144 + k)
    for (int i = tid; i < 144 * 32; i += 128) {
        int k = i >> 5, n = i & 31;
        Bs[k * 33 + n] = w2[n * 144 + k];
    }
    __syncthreads();

    int wave = tid >> 5, lane = tid & 31;
    int l = lane & 15;
    int koff = (lane >> 4) << 1;  // 0 or 2 (K half per A/B layout)
    const float* Arow = &As[(wave * 16 + l) * 148];

#pragma unroll
    for (int nt = 0; nt < 2; ++nt) {
        v8f acc = {};
#pragma unroll 4
        for (int k = 0; k < 144; k += 4) {
            v2f a, b;
            a.x = Arow[k + koff];
            a.y = Arow[k + koff + 1];
            b.x = Bs[(k + koff) * 33 + nt * 16 + l];
            b.y = Bs[(k + koff + 1) * 33 + nt * 16 + l];
            acc = wmma4(a, b, acc);
        }
        int n = nt * 16 + l;
        float bn = b2[n];
        int rbase = (lane >> 4) * 8;  // C/D layout: rows v / v+8
#pragma unroll
        for (int v = 0; v < 8; ++v) {
            int g = m_base + wave * 16 + rbase + v;
            int b = g / 196;
            int p = g - b * 196;
            int oy = p / 14;
            int ox = p - oy * 14;
            float val = fmaxf(fpq(acc[v] + bn), 0.0f);
            h2[((b * 32 + n) * 14 + oy) * 14 + ox] = val;
        }
    }
}

// ---------------------------------------------------------------------------
// Kernel 3: maxpool2 + fpq -> h3 [4096,32,7,7]
// ---------------------------------------------------------------------------
__global__ __launch_bounds__(256) void pool2_kernel(
    const float* __restrict__ h2, float* __restrict__ h3) {
    int idx = blockIdx.x * blockDim.x + threadIdx.x;
    const int total = 4096 * 32 * 7 * 7;
    if (idx >= total) return;
    int px = idx % 7;
    int py = (idx / 7) % 7;
    int c = (idx / 49) % 32;
    int b = idx / (49 * 32);
    const float* base = h2 + ((b * 32 + c) * 14 + 2 * py) * 14 + 2 * px;
    float m = fmaxf(fmaxf(base[0], base[1]), fmaxf(base[14], base[15]));
    h3[idx] = fpq(m);
}

// ---------------------------------------------------------------------------
// Kernel 4: fc1 GEMM  [4096 x 1568] x [1568 x 128] via WMMA f32, fused
// bias+fpq+relu. One wave per 16x16 output tile (2048 tiles, 392 WMMA each).
// NCHW flatten of [b][32][7][7] is naturally row-major [4096][1568].
// ---------------------------------------------------------------------------
__global__ __launch_bounds__(128) void fc1_wmma_kernel(
    const float* __restrict__ h3, const float* __restrict__ w,
    const float* __restrict__ bias, float* __restrict__ h4) {
    int tid = threadIdx.x;
    int wave = tid >> 5, lane = tid & 31;
    int wid = blockIdx.x * 4 + wave;  // 0..2047
    int mtile = wid >> 3;             // 0..255
    int ntile = wid & 7;              // 0..7
    int l = lane & 15;
    int koff = (lane >> 4) << 1;
    int m0 = mtile * 16;
    int n = ntile * 16 + l;
    const float* Arow = h3 + (m0 + l) * 1568;
    const float* Bcol = w + n * 1568;  // B[k][n] = fc1_w[n][k]
    v8f acc = {};
#pragma unroll 4
    for (int k = 0; k < 1568; k += 4) {
        v2f a, b;
        a.x = Arow[k + koff];
        a.y = Arow[k + koff + 1];
        b.x = Bcol[k + koff];
        b.y = Bcol[k + koff + 1];
        acc = wmma4(a, b, acc);
    }
    float bn = bias[n];
    int rbase = (lane >> 4) * 8;
#pragma unroll
    for (int v = 0; v < 8; ++v) {
        int row = m0 + rbase + v;
        h4[row * 128 + n] = fmaxf(fpq(acc[v] + bn), 0.0f);
    }
}

// ---------------------------------------------------------------------------
// Kernel 5: fc2 GEMM [4096 x 128] x [128 x 10] via WMMA f32 (N padded to 16,
// masked columns so EXEC stays all-1s). Fused bias + fpq (no relu).
// ---------------------------------------------------------------------------
__global__ __launch_bounds__(128) void fc2_wmma_kernel(
    const float* __restrict__ h4, const float* __restrict__ w,
    const float* __restrict__ bias, float* __restrict__ out) {
    int tid = threadIdx.x;
    int wave = tid >> 5, lane = tid & 31;
    int mtile = blockIdx.x * 4 + wave;  // 0..255
    int l = lane & 15;
    int koff = (lane >> 4) << 1;
    int m0 = mtile * 16;
    bool nvalid = (l < 10);
    int nc = nvalid ? l : 9;
    float nmask = nvalid ? 1.0f : 0.0f;
    const float* Arow = h4 + (m0 + l) * 128;
    const float* Bcol = w + nc * 128;
    v8f acc = {};
#pragma unroll 8
    for (int k = 0; k < 128; k += 4) {
        v2f a, b;
        a.x = Arow[k + koff];
        a.y = Arow[k + koff + 1];
        b.x = Bcol[k + koff] * nmask;   // zero padded N columns
        b.y = Bcol[k + koff + 1] * nmask;
        acc = wmma4(a, b, acc);
    }
    float bn = nvalid ? bias[l] : 0.0f;
    int rbase = (lane >> 4) * 8;
#pragma unroll
    for (int v = 0; v < 8; ++v) {
        int row = m0 + rbase + v;
        float d = fpq(acc[v] + bn);
        if (nvalid) out[row * 10 + l] = d;
    }
}

extern "C" void kernel_launch(void* const* d_in, const int* in_sizes, int n_in,
                              void* d_out, int out_size, void* d_ws, size_t ws_size,
                              hipStream_t stream) {
    const float* x   = (const float*)d_in[0];
    const float* c1w = (const float*)d_in[1];
    const float* c1b = (const float*)d_in[2];
    const float* c2w = (const float*)d_in[3];
    const float* c2b = (const float*)d_in[4];
    const float* f1w = (const float*)d_in[5];
    const float* f1b = (const float*)d_in[6];
    const float* f2w = (const float*)d_in[7];
    const float* f2b = (const float*)d_in[8];

    char* ws = (char*)d_ws;
    float* h1 = (float*)(ws);                  // [4096,16,14,14]  51,380,224 B
    float* h2 = (float*)(ws + 51380224);       // [4096,32,14,14] 102,760,448 B
    float* h3 = (float*)(ws + 154140672);      // [4096,32,7,7]    25,690,112 B
    float* h4 = (float*)(ws + 179830784);      // [4096,128]        2,097,152 B
    float* out = (float*)d_out;                // [4096,10]

    conv1_pool_kernel<<<50176, 256, 0, stream>>>(x, c1w, c1b, h1);
    conv2_wmma_kernel<<<12544, 128, 0, stream>>>(h1, c2w, c2b, h2);
    pool2_kernel<<<25088, 256, 0, stream>>>(h2, h3);
    fc1_wmma_kernel<<<512, 128, 0, stream>>>(h3, f1w, f1b, h4);
    fc2_wmma_kernel<<<64, 128, 0, stream>>>(h4, f2w, f2b, out);
}